// SampleModel_77610059038911
// MI455X (gfx1250) — compile-verified
//
#include <hip/hip_runtime.h>
#include <math.h>

// ---- problem constants (match reference) ----
#define N_PTS   131072
#define DIM     256
#define K_CENT  256
#define INV_T   10.0f        // 1/TEMP
#define NBLK_FPS 16384       // N_PTS / 8 rows-per-block
#define NBLK_GEMM 1024       // N_PTS / 128 rows-per-block

typedef float v2f __attribute__((ext_vector_type(2)));
typedef float v8f __attribute__((ext_vector_type(8)));

__device__ __forceinline__ v8f wmma_f32_4(v2f a, v2f b, v8f c) {
  // D = A(16x4,f32) * B(4x16,f32) + C ; fp32 matrix path on gfx1250
  return __builtin_amdgcn_wmma_f32_16x16x4_f32(
      /*neg_a=*/false, a, /*neg_b=*/false, b,
      /*c_mod=*/(short)0, c, /*reuse_a=*/false, /*reuse_b=*/false);
}

// CDNA5 async global->LDS copy (ASYNCcnt-tracked), per CDNA5 ISA
__device__ __forceinline__ void async_g2l_b128(unsigned lds_byte_addr, const void* gaddr) {
  asm volatile("global_load_async_to_lds_b128 %0, %1, off"
               :: "v"(lds_byte_addr), "v"((unsigned long long)(size_t)gaddr)
               : "memory");
}
__device__ __forceinline__ void async_g2l_b32(unsigned lds_byte_addr, const void* gaddr) {
  asm volatile("global_load_async_to_lds_b32 %0, %1, off"
               :: "v"(lds_byte_addr), "v"((unsigned long long)(size_t)gaddr)
               : "memory");
}
__device__ __forceinline__ void wait_async0() {
  asm volatile("s_wait_asynccnt 0x0" ::: "memory");
}
__device__ __forceinline__ unsigned lds_addr_of(const void* p) {
  // generic shared pointer: low 32 bits are the LDS byte address
  return (unsigned)(size_t)p;
}

__device__ __forceinline__ float dot8(float4 a0, float4 a1, float4 b0, float4 b1) {
  float s = a0.x * b0.x;
  s = fmaf(a0.y, b0.y, s);
  s = fmaf(a0.z, b0.z, s);
  s = fmaf(a0.w, b0.w, s);
  s = fmaf(a1.x, b1.x, s);
  s = fmaf(a1.y, b1.y, s);
  s = fmaf(a1.z, b1.z, s);
  s = fmaf(a1.w, b1.w, s);
  return s;
}

// ---------------- Phase A: row squared norms + init ----------------
__global__ __launch_bounds__(256) void row_norm2_init(
    const float* __restrict__ f, float* __restrict__ r2,
    float* __restrict__ dist, int* __restrict__ ids, int* __restrict__ cur) {
  int t = threadIdx.x, wid = t >> 5, lane = t & 31;
  int row = blockIdx.x * 8 + wid;
  const float4* xr = (const float4*)(f + (size_t)row * DIM);
  float4 a0 = xr[lane * 2 + 0];
  float4 a1 = xr[lane * 2 + 1];
  float s = dot8(a0, a1, a0, a1);
#pragma unroll
  for (int off = 16; off > 0; off >>= 1) s += __shfl_xor(s, off, 32);
  if (lane == 0) {
    r2[row]   = s;
    dist[row] = 3.0e38f;   // reference's 1e20 sentinel (squared domain)
  }
  if (blockIdx.x == 0 && t == 0) { ids[0] = 0; *cur = 0; }
}

// ---------------- Phase B1: FPS distance update + per-block argmax -------
__global__ __launch_bounds__(256) void fps_step(
    const float* __restrict__ f, const float* __restrict__ r2,
    float* __restrict__ dist, const int* __restrict__ cur,
    float* __restrict__ pval, int* __restrict__ pidx) {
  __shared__ __align__(16) float pbuf[DIM];
  __shared__ float wval[8];
  __shared__ int   widx[8];
  int t = threadIdx.x, wid = t >> 5, lane = t & 31;
  int nid = *cur;
  // async broadcast of the newly selected point into LDS
  async_g2l_b32(lds_addr_of(&pbuf[t]), f + (size_t)nid * DIM + t);
  wait_async0();
  __syncthreads();

  int row = blockIdx.x * 8 + wid;
  const float4* xr = (const float4*)(f + (size_t)row * DIM);
  const float4* pr = (const float4*)pbuf;
  float4 x0 = xr[lane * 2 + 0], x1 = xr[lane * 2 + 1];
  float4 p0 = pr[lane * 2 + 0], p1 = pr[lane * 2 + 1];
  float dot = dot8(x0, x1, p0, p1);
#pragma unroll
  for (int off = 16; off > 0; off >>= 1) dot += __shfl_xor(dot, off, 32);

  if (lane == 0) {
    float d = r2[row] - 2.0f * dot + r2[nid];
    d = fmaxf(d, 0.0f);
    float nd = fminf(dist[row], d);
    dist[row] = nd;
    wval[wid] = nd;
    widx[wid] = row;
  }
  __syncthreads();
  if (t == 0) {
    float bv = wval[0]; int bi = widx[0];
#pragma unroll
    for (int i = 1; i < 8; ++i) {
      if (wval[i] > bv || (wval[i] == bv && widx[i] < bi)) { bv = wval[i]; bi = widx[i]; }
    }
    pval[blockIdx.x] = bv;
    pidx[blockIdx.x] = bi;
  }
}

// ---------------- Phase B2: global argmax over block partials ------------
__global__ __launch_bounds__(1024) void fps_reduce(
    const float* __restrict__ pval, const int* __restrict__ pidx,
    int* __restrict__ ids, int* __restrict__ cur, int iter) {
  __shared__ float sv[1024];
  __shared__ int   si[1024];
  int t = threadIdx.x;
  float bv = -1.0f; int bi = 0x7fffffff;
  for (int i = t; i < NBLK_FPS; i += 1024) {
    float v = pval[i]; int ix = pidx[i];
    if (v > bv || (v == bv && ix < bi)) { bv = v; bi = ix; }
  }
  sv[t] = bv; si[t] = bi;
  __syncthreads();
  for (int s = 512; s > 0; s >>= 1) {
    if (t < s) {
      if (sv[t + s] > sv[t] || (sv[t + s] == sv[t] && si[t + s] < si[t])) {
        sv[t] = sv[t + s]; si[t] = si[t + s];
      }
    }
    __syncthreads();
  }
  if (t == 0) { ids[iter + 1] = si[0]; *cur = si[0]; }
}

// ---------------- Phase C: gather + L2-normalize centroids ----------------
__global__ __launch_bounds__(256) void build_c(
    const float* __restrict__ f, const float* __restrict__ r2,
    const int* __restrict__ ids, float* __restrict__ cmat) {
  int b = blockIdx.x;                 // centroid 0..255
  int id = ids[b];
  float nrm = fmaxf(sqrtf(r2[id]), 1e-12f);
  cmat[(size_t)b * DIM + threadIdx.x] = f[(size_t)id * DIM + threadIdx.x] / nrm;
}

// ---------------- Phase E: cent_prod_exp = exp(c @ c^T / T) (WMMA) --------
__global__ __launch_bounds__(256) void cent_exp(
    const float* __restrict__ cmat, float* __restrict__ cexp) {
  int t = threadIdx.x, wid = t >> 5, lane = t & 31;
  int lm = lane & 15, lh = lane >> 4;
  int tile = blockIdx.x * 8 + wid;    // 0..255 tiles over 256x256 output
  int ti = tile >> 4, tj = tile & 15;
  const float2* arow = (const float2*)(cmat + (size_t)(ti * 16 + lm) * DIM);
  const float2* brow = (const float2*)(cmat + (size_t)(tj * 16 + lm) * DIM);
  v8f acc = {};
#pragma unroll 8
  for (int k2 = 0; k2 < DIM / 2; k2 += 2) {   // K-step of 4
    float2 af = arow[k2 + lh];
    float2 bf = brow[k2 + lh];
    v2f a = {af.x, af.y};
    v2f b = {bf.x, bf.y};
    acc = wmma_f32_4(a, b, acc);
  }
#pragma unroll
  for (int j = 0; j < 8; ++j) {
    int m = ti * 16 + j + 8 * lh;     // C layout: row = vgpr + 8*(lane/16)
    int n = tj * 16 + lm;             //           col = lane%16
    cexp[(size_t)m * K_CENT + n] = expf(acc[j] * INV_T);
  }
}

// ---------------- Phase E2: deterministic column sums ---------------------
__global__ __launch_bounds__(256) void cent_sum(
    const float* __restrict__ cexp, float* __restrict__ csum) {
  int t = threadIdx.x;                // column j
  float s = 0.0f;
  for (int i = 0; i < K_CENT; ++i) s += cexp[(size_t)i * K_CENT + t];
  csum[t] = s;
}

// ---- Phase D: fused GEMM (features @ c^T) + row max/argmax + J (WMMA) ----
// Block = 256 threads = 8 waves, owns 128 rows. A stripe (128 KB) staged once
// into LDS via async copies; each 16-row B tile (16 KB) staged per iteration.
// 145 KB LDS/block -> 2 blocks per 320 KB WGP; all tile reuse served by LDS.
__global__ __launch_bounds__(256) void logits_max_J(
    const float* __restrict__ f, const float* __restrict__ cmat,
    const float* __restrict__ csum, float* __restrict__ jpart) {
  __shared__ __align__(16) float aTile[128 * DIM];   // 128 KB
  __shared__ __align__(16) float bTile[16 * DIM];    // 16 KB
  __shared__ float red[256];
  int t = threadIdx.x, wid = t >> 5, lane = t & 31;
  int lm = lane & 15, lh = lane >> 4;

  // stage A: 128 feature rows for this block (async, ASYNCcnt-tracked)
  {
    unsigned ldsA = lds_addr_of(&aTile[0]);
    const float4* src = (const float4*)(f + (size_t)blockIdx.x * 128 * DIM);
#pragma unroll
    for (int i = 0; i < 32; ++i) {
      int idx = t + 256 * i;          // float4 index into 128x256 tile
      async_g2l_b128(ldsA + (unsigned)idx * 16u, src + idx);
    }
  }

  float rmax[8];
  int   rarg[8];
#pragma unroll
  for (int j = 0; j < 8; ++j) { rmax[j] = -3.4e38f; rarg[j] = 0; }

  unsigned ldsB = lds_addr_of(&bTile[0]);
  const float2* arow = (const float2*)(aTile + (size_t)(wid * 16 + lm) * DIM);
  const float2* brow = (const float2*)(bTile + (size_t)lm * DIM);

  for (int nt = 0; nt < 16; ++nt) {                  // 16 centroid tiles
    // stage B: 16 centroid rows (async)
    const float4* bsrc = (const float4*)(cmat + (size_t)nt * 16 * DIM);
#pragma unroll
    for (int i = 0; i < 4; ++i) {
      int idx = t + 256 * i;
      async_g2l_b128(ldsB + (unsigned)idx * 16u, bsrc + idx);
    }
    wait_async0();
    __syncthreads();                  // A (first iter) + B visible to all

    v8f acc = {};
#pragma unroll 8
    for (int k2 = 0; k2 < DIM / 2; k2 += 2) {        // 64 WMMA K-steps, LDS-fed
      float2 af = arow[k2 + lh];
      float2 bf = brow[k2 + lh];
      v2f a = {af.x, af.y};
      v2f b = {bf.x, bf.y};
      acc = wmma_f32_4(a, b, acc);
    }
    __syncthreads();                  // all waves done with bTile before rewrite

    int col = nt * 16 + lm;
#pragma unroll
    for (int j = 0; j < 8; ++j) {
      if (acc[j] > rmax[j]) { rmax[j] = acc[j]; rarg[j] = col; }  // strict >: first-index tie
    }
  }

  // cross-lane argmax within each half-wave (lanes of a half share rows)
#pragma unroll
  for (int off = 1; off < 16; off <<= 1) {
#pragma unroll
    for (int j = 0; j < 8; ++j) {
      float ov = __shfl_xor(rmax[j], off, 32);
      int   oa = __shfl_xor(rarg[j], off, 32);
      if (ov > rmax[j] || (ov == rmax[j] && oa < rarg[j])) { rmax[j] = ov; rarg[j] = oa; }
    }
  }

  float localJ = 0.0f;
  if (lm == 0) {      // lanes 0 and 16 each own 8 distinct rows
#pragma unroll
    for (int j = 0; j < 8; ++j) {
      float p = expf(rmax[j] * INV_T);          // reproduces fp32 inf behavior
      float s = csum[rarg[j]];                  // BALANCE == 1
      localJ += logf(p) - logf(p + s);
    }
  }
  red[t] = localJ;
  __syncthreads();
  for (int s = 128; s > 0; s >>= 1) {
    if (t < s) red[t] += red[t + s];
    __syncthreads();
  }
  if (t == 0) jpart[blockIdx.x] = red[0];
}

// ---------------- Phase F: final deterministic reduction ------------------
__global__ __launch_bounds__(256) void finalize(
    const float* __restrict__ jpart, float* __restrict__ out) {
  __shared__ float red[256];
  int t = threadIdx.x;
  float s = 0.0f;
  for (int i = t; i < NBLK_GEMM; i += 256) s += jpart[i];
  red[t] = s;
  __syncthreads();
  for (int k = 128; k > 0; k >>= 1) {
    if (t < k) red[t] += red[t + k];
    __syncthreads();
  }
  if (t == 0) out[0] = -red[0] / (float)N_PTS;
}

extern "C" void kernel_launch(void* const* d_in, const int* in_sizes, int n_in,
                              void* d_out, int out_size, void* d_ws, size_t ws_size,
                              hipStream_t stream) {
  const float* f = (const float*)d_in[0];
  float* out = (float*)d_out;
  char* ws = (char*)d_ws;

  // workspace layout (bytes)
  float* r2   = (float*)(ws + 0);          // 131072 * 4
  float* dist = (float*)(ws + 524288);     // 131072 * 4
  float* cmat = (float*)(ws + 1048576);    // 256*256*4
  float* cexp = (float*)(ws + 1310720);    // 256*256*4
  float* csum = (float*)(ws + 1572864);    // 256*4
  int*   ids  = (int*)  (ws + 1573888);    // 256*4
  int*   cur  = (int*)  (ws + 1574912);    // control
  float* pval = (float*)(ws + 1574976);    // 16384*4
  int*   pidx = (int*)  (ws + 1640512);    // 16384*4
  float* jprt = (float*)(ws + 1706048);    // 1024*4

  // A: norms + init
  row_norm2_init<<<NBLK_FPS, 256, 0, stream>>>(f, r2, dist, ids, cur);

  // B: 255 sequential FPS steps (L2-resident passes over features)
  for (int it = 0; it < K_CENT - 1; ++it) {
    fps_step<<<NBLK_FPS, 256, 0, stream>>>(f, r2, dist, cur, pval, pidx);
    fps_reduce<<<1, 1024, 0, stream>>>(pval, pidx, ids, cur, it);
  }

  // C: centroids
  build_c<<<K_CENT, 256, 0, stream>>>(f, r2, ids, cmat);

  // E: exp(c @ c^T / T) and column sums
  cent_exp<<<32, 256, 0, stream>>>(cmat, cexp);
  cent_sum<<<1, 256, 0, stream>>>(cexp, csum);

  // D: fused big GEMM + row max/argmax + per-row J (LDS-staged, async copies)
  logits_max_J<<<NBLK_GEMM, 256, 0, stream>>>(f, cmat, csum, jprt);

  // F: scalar loss
  finalize<<<1, 256, 0, stream>>>(jprt, out);
}